// GCN_29927332118584
// MI455X (gfx1250) — compile-verified
//
#include <hip/hip_runtime.h>

typedef float v2f __attribute__((ext_vector_type(2)));
typedef float v8f __attribute__((ext_vector_type(8)));

#define DD 64

// ---------------------------------------------------------------------------
// Elementwise / edge kernels
// ---------------------------------------------------------------------------

__global__ void __launch_bounds__(256)
fill_f32(float* __restrict__ p, float v, int n) {
  int i = blockIdx.x * blockDim.x + threadIdx.x;
  if (i < n) p[i] = v;
}

__global__ void __launch_bounds__(256)
degree_count(const int* __restrict__ dst, float* __restrict__ deg, int ne) {
  int e = blockIdx.x * blockDim.x + threadIdx.x;
  if (e < ne) unsafeAtomicAdd(&deg[dst[e]], 1.0f);
}

__global__ void __launch_bounds__(256)
rsqrt_inplace(float* __restrict__ p, int n) {
  int i = blockIdx.x * blockDim.x + threadIdx.x;
  if (i < n) p[i] = rsqrtf(p[i]);   // deg >= 1 always (self-loops folded in)
}

// acc[i,c] = dinv[i]^2 * xw[i,c]   (self-loop message, initializes accumulator)
__global__ void __launch_bounds__(256)
selfloop_init(const float* __restrict__ xw, const float* __restrict__ dinv,
              float* __restrict__ acc, int n) {
  int t = blockIdx.x * blockDim.x + threadIdx.x;
  if (t >= n * DD) return;
  int i = t >> 6;
  float di = dinv[i];
  acc[t] = di * di * xw[t];
}

// acc[dst,c] += msg[src,c] * dinv[src]*dinv[dst]   (64 lanes per edge, coalesced)
__global__ void __launch_bounds__(256)
scatter_edges(const int* __restrict__ src, const int* __restrict__ dst,
              const float* __restrict__ msg, const float* __restrict__ dinv,
              float* __restrict__ acc, int ne) {
  int t = blockIdx.x * blockDim.x + threadIdx.x;
  int e = t >> 6;
  if (e >= ne) return;
  int c = t & 63;
  int s = src[e];
  int d = dst[e];
  float nrm = dinv[s] * dinv[d];
  unsafeAtomicAdd(&acc[(size_t)d * DD + c], msg[(size_t)s * DD + c] * nrm);
}

__global__ void __launch_bounds__(256)
bias_relu_inplace(float* __restrict__ h, const float* __restrict__ b, int n) {
  int t = blockIdx.x * blockDim.x + threadIdx.x;
  if (t >= n * DD) return;
  h[t] = fmaxf(h[t] + b[t & 63], 0.0f);
}

// out[i,c] = out[i,c](=x@Wemb) + bemb[c] + b2[c] + dinv[i]^2 * hw2[i,c]
__global__ void __launch_bounds__(256)
final_combine(float* __restrict__ out, const float* __restrict__ hw2,
              const float* __restrict__ dinv, const float* __restrict__ b2,
              const float* __restrict__ bemb, int n) {
  int t = blockIdx.x * blockDim.x + threadIdx.x;
  if (t >= n * DD) return;
  int i = t >> 6, c = t & 63;
  float di = dinv[i];
  out[t] = out[t] + bemb[c] + b2[c] + di * di * hw2[t];
}

// ---------------------------------------------------------------------------
// FP32 WMMA GEMM:  C[nrows x 64] = A[nrows x 64] * W[64 x 64]
// One wave computes a 16x64 output tile with V_WMMA_F32_16X16X4_F32.
// A-frag (16x4 f32): lanes 0-15 rows M=0..15 holding K = kb,kb+1;
//                    lanes 16-31 same rows holding K = kb+2,kb+3.
// B-frag (4x16 f32): lane -> col N = lane%16; VGPR0/1 hold rows K = kb(+2), kb+1(+3).
// C/D (16x16 f32):   VGPR r -> row r (lanes 0-15) / r+8 (lanes 16-31), col = lane%16.
// ---------------------------------------------------------------------------
__global__ void __launch_bounds__(256)
gemm_n64_wmma(const float* __restrict__ A, const float* __restrict__ W,
              float* __restrict__ C, int nrows) {
  const int lane = threadIdx.x & 31;
  const int wave = threadIdx.x >> 5;
  const int tile = blockIdx.x * 8 + wave;          // 16-row tile index
  const int row0 = tile * 16;
  if (row0 >= nrows) return;                       // wave-uniform exit

  const int half = lane >> 4;                      // 0: lanes 0-15, 1: 16-31
  const int l16  = lane & 15;

  // Clamp the A read row so partial tiles keep EXEC all-ones for WMMA.
  int ar = row0 + l16;
  if (ar >= nrows) ar = nrows - 1;
  const float* __restrict__ arow = A + (size_t)ar * DD;

  v8f acc0 = {}, acc1 = {}, acc2 = {}, acc3 = {};

#pragma unroll
  for (int k = 0; k < 16; ++k) {
    const int kb = k * 4 + half * 2;               // this lane-half's K base
    v2f a = *(const v2f*)(arow + kb);              // K = kb, kb+1 (8B aligned)

    const float* __restrict__ wr0 = W + (size_t)kb * DD;        // row K=kb
    const float* __restrict__ wr1 = W + (size_t)(kb + 1) * DD;  // row K=kb+1
    v2f b0, b1, b2, b3;
    b0.x = wr0[l16];      b0.y = wr1[l16];
    b1.x = wr0[16 + l16]; b1.y = wr1[16 + l16];
    b2.x = wr0[32 + l16]; b2.y = wr1[32 + l16];
    b3.x = wr0[48 + l16]; b3.y = wr1[48 + l16];

    acc0 = __builtin_amdgcn_wmma_f32_16x16x4_f32(false, a, false, b0, (short)0, acc0, false, false);
    acc1 = __builtin_amdgcn_wmma_f32_16x16x4_f32(false, a, false, b1, (short)0, acc1, false, false);
    acc2 = __builtin_amdgcn_wmma_f32_16x16x4_f32(false, a, false, b2, (short)0, acc2, false, false);
    acc3 = __builtin_amdgcn_wmma_f32_16x16x4_f32(false, a, false, b3, (short)0, acc3, false, false);
  }

  float* __restrict__ crow = C + (size_t)row0 * DD;
#pragma unroll
  for (int r = 0; r < 8; ++r) {
    const int rr = r + half * 8;
    if (row0 + rr < nrows) {
      crow[(size_t)rr * DD + l16]      = acc0[r];
      crow[(size_t)rr * DD + 16 + l16] = acc1[r];
      crow[(size_t)rr * DD + 32 + l16] = acc2[r];
      crow[(size_t)rr * DD + 48 + l16] = acc3[r];
    }
  }
}

// ---------------------------------------------------------------------------
// Launch
// ---------------------------------------------------------------------------
extern "C" void kernel_launch(void* const* d_in, const int* in_sizes, int n_in,
                              void* d_out, int out_size, void* d_ws, size_t ws_size,
                              hipStream_t stream) {
  const float* x    = (const float*)d_in[0];
  const int*   ei   = (const int*)d_in[1];
  const float* W1   = (const float*)d_in[2];
  const float* b1   = (const float*)d_in[3];
  const float* W2   = (const float*)d_in[4];
  const float* b2   = (const float*)d_in[5];
  const float* Wemb = (const float*)d_in[6];
  const float* bemb = (const float*)d_in[7];
  float* out = (float*)d_out;

  const int N = in_sizes[0] / DD;
  const int E = in_sizes[1] / 2;
  const int* src = ei;
  const int* dst = ei + E;

  // Workspace layout (floats): dinv[Npad] | buf1[Npad*64] | buf2[Npad*64]
  float* ws = (float*)d_ws;
  const size_t npad = ((size_t)N + 63) & ~(size_t)63;
  float* dinv = ws;
  float* buf1 = ws + npad;                 // xW1, later hW2
  float* buf2 = buf1 + npad * DD;          // h accumulator, later h

  const int B   = 256;
  const int gN  = (N + B - 1) / B;
  const int gE  = (E + B - 1) / B;
  const int gNE = (N * DD + B - 1) / B;
  const int gEE = (int)(((size_t)E * DD + (size_t)B - 1) / (size_t)B);
  const int gG  = (((N + 15) / 16) + 7) / 8;   // 8 waves (16-row tiles) per block

  // degrees (self-loop => start at 1.0), then deg^-1/2
  fill_f32<<<gN, B, 0, stream>>>(dinv, 1.0f, N);
  degree_count<<<gE, B, 0, stream>>>(dst, dinv, E);
  rsqrt_inplace<<<gN, B, 0, stream>>>(dinv, N);

  // layer 1: h = relu(scatter(norm * (x@W1)) + b1)
  gemm_n64_wmma<<<gG, B, 0, stream>>>(x, W1, buf1, N);
  selfloop_init<<<gNE, B, 0, stream>>>(buf1, dinv, buf2, N);
  scatter_edges<<<gEE, B, 0, stream>>>(src, dst, buf1, dinv, buf2, E);
  bias_relu_inplace<<<gNE, B, 0, stream>>>(buf2, b1, N);

  // layer 2 + residual: out = scatter(norm * (h@W2)) + b2 + x@Wemb + bemb
  gemm_n64_wmma<<<gG, B, 0, stream>>>(buf2, W2, buf1, N);     // hW2
  gemm_n64_wmma<<<gG, B, 0, stream>>>(x, Wemb, out, N);       // emb -> out
  final_combine<<<gNE, B, 0, stream>>>(out, buf1, dinv, b2, bemb, N);
  scatter_edges<<<gEE, B, 0, stream>>>(src, dst, buf1, dinv, out, E);
}